// Denoiser_73598559584966
// MI455X (gfx1250) — compile-verified
//
#include <hip/hip_runtime.h>
#include <math.h>

// Self-attention (Q=K=V), B=4, N=4096, C=64, fp32, flash-style streaming.
// Matrix path: V_WMMA_F32_16X16X4_F32 (fp32-exact vs the fp32 reference).
// Trick: compute S^T = K*Q^T and out^T = V^T*P^T so the softmax axis lands
// down the VGPRs of the C-layout -> reductions are in-lane + one xor-16
// shuffle, softmax stats are per-lane scalars, and P^T feeds the second
// WMMA directly (4 shuffles, no LDS bounce).

typedef __attribute__((ext_vector_type(2))) float v2f;
typedef __attribute__((ext_vector_type(4))) float v4f;
typedef __attribute__((ext_vector_type(8))) float v8f;

#define NTOK 4096        // D*H*W = 16*16*16
#define CDIM 64
#define NB   4
#define BK   128         // K/V rows staged in LDS per outer iteration
#define WAVES 4          // waves per workgroup
#define ROWS_PER_WG (WAVES * 16)   // 64 query rows per workgroup

__global__ __launch_bounds__(WAVES * 32)
void attn_f32_wmma_kernel(const float* __restrict__ x, float* __restrict__ out) {
    __shared__ float kv[BK * CDIM];            // 32 KB staged K/V rows (row-major)

    const int lane = threadIdx.x & 31;
    const int wave = threadIdx.x >> 5;
    const int ln   = lane & 15;                // query row within wave tile
    const int hi   = lane >> 4;                // lane half: 0 or 1
    const int b    = blockIdx.y;
    const int i0   = blockIdx.x * ROWS_PER_WG + wave * 16;

    const float* xb = x + (size_t)b * NTOK * CDIM;

    // Q fragments. Used as B-matrix of S^T = K * Q^T:
    // B-layout (4x16): lane ln = column i, v = {K-rows 2hi, 2hi+1} of subblock.
    // B[c][i] = Q[i][c]  ->  identical addresses to an A-layout load of Q.
    v2f qa[CDIM / 4];
#pragma unroll
    for (int kb = 0; kb < CDIM / 4; ++kb)
        qa[kb] = *(const v2f*)(xb + (size_t)(i0 + ln) * CDIM + kb * 4 + 2 * hi);

    // acc = out^T in C-layout per 16-col tile ct:
    // acc[ct][r] @ lane(ln,hi) = out[i0+ln][ct*16 + r + 8*hi]
    v8f acc[4] = {};
    float m = -INFINITY;    // per-lane: stats of query row i0+ln
    float l = 0.0f;

#pragma unroll 1
    for (int blk = 0; blk < NTOK; blk += BK) {
        __syncthreads();   // all waves done reading previous kv block
        {   // cooperative stage: BK*CDIM floats = 2048 float4, 128 threads
            const v4f* src = (const v4f*)(xb + (size_t)blk * CDIM);
            v4f*       dst = (v4f*)kv;
#pragma unroll
            for (int k = 0; k < (BK * CDIM / 4) / (WAVES * 32); ++k)
                dst[threadIdx.x + k * (WAVES * 32)] = src[threadIdx.x + k * (WAVES * 32)];
        }
        __syncthreads();

#pragma unroll 1
        for (int jt = 0; jt < BK; jt += 16) {
            // ---- S^T = K * Q^T.  A = K tile (A-layout: lane ln = j-row jt+ln,
            // v = {c,c+1} at c = 4kb+2hi), B = qa.  C-layout result:
            // s[r] @ lane(ln,hi) = S[query i0+ln][key jt + r + 8*hi]
            v8f s = {};
#pragma unroll
            for (int kb = 0; kb < CDIM / 4; ++kb) {
                v2f kf = *(const v2f*)&kv[(jt + ln) * CDIM + kb * 4 + 2 * hi];
                s = __builtin_amdgcn_wmma_f32_16x16x4_f32(
                        false, kf, false, qa[kb], (short)0, s, false, false);
            }

            // ---- online softmax over j: 8 in-lane values + xor-16 partner
            float tmax = fmaxf(fmaxf(fmaxf(s[0], s[1]), fmaxf(s[2], s[3])),
                               fmaxf(fmaxf(s[4], s[5]), fmaxf(s[6], s[7])));
            tmax = fmaxf(tmax, __shfl_xor(tmax, 16, 32));
            float mnew = fmaxf(m, tmax);
            float sc = __expf(m - mnew);          // m = -inf first pass -> 0
            float pr[8];
            float tsum = 0.0f;
#pragma unroll
            for (int r = 0; r < 8; ++r) {
                pr[r] = __expf(s[r] - mnew);      // pr[r] = P[ln][jt + r + 8hi]
                tsum += pr[r];
            }
            tsum += __shfl_xor(tsum, 16, 32);
            l = l * sc + tsum;
            m = mnew;
#pragma unroll
            for (int ct = 0; ct < 4; ++ct)
                acc[ct] *= sc;                    // per-lane scalar rescale

            // ---- build B-fragments of P^T (== A-layout contents of P):
            // frag kb needs P[ln][4kb+2hi], P[ln][4kb+2hi+1]; half the values
            // come from the xor-16 partner lane.
            float t0 = hi ? pr[0] : pr[2];
            float t1 = hi ? pr[1] : pr[3];
            float t2 = hi ? pr[4] : pr[6];
            float t3 = hi ? pr[5] : pr[7];
            float u0 = __shfl_xor(t0, 16, 32);
            float u1 = __shfl_xor(t1, 16, 32);
            float u2 = __shfl_xor(t2, 16, 32);
            float u3 = __shfl_xor(t3, 16, 32);
            v2f pa[4];
            pa[0].x = hi ? u0 : pr[0];  pa[0].y = hi ? u1 : pr[1];
            pa[1].x = hi ? u2 : pr[4];  pa[1].y = hi ? u3 : pr[5];
            pa[2].x = hi ? pr[2] : u0;  pa[2].y = hi ? pr[3] : u1;
            pa[3].x = hi ? pr[6] : u2;  pa[3].y = hi ? pr[7] : u3;

            // ---- out^T += V^T * P^T.  A = V^T tile ct (A-layout: lane ln =
            // c-row ct*16+ln, v = {j, j+1} at j = jt+4kb+2hi), B = pa.
#pragma unroll
            for (int ct = 0; ct < 4; ++ct) {
#pragma unroll
                for (int kb = 0; kb < 4; ++kb) {
                    v2f vt;
                    vt.x = kv[(jt + kb * 4 + 2 * hi)     * CDIM + ct * 16 + ln];
                    vt.y = kv[(jt + kb * 4 + 2 * hi + 1) * CDIM + ct * 16 + ln];
                    acc[ct] = __builtin_amdgcn_wmma_f32_16x16x4_f32(
                                  false, vt, false, pa[kb], (short)0, acc[ct], false, false);
                }
            }
        }
    }

    // ---- normalize (per-lane l) and store: lane(ln,hi) owns 8 consecutive
    // columns ct*16 + 8hi .. +7 of row i0+ln -> two b128 stores per tile.
    const float inv = 1.0f / l;
    float* ob = out + (size_t)b * NTOK * CDIM + (size_t)(i0 + ln) * CDIM + 8 * hi;
#pragma unroll
    for (int ct = 0; ct < 4; ++ct) {
        v4f lo = { acc[ct][0] * inv, acc[ct][1] * inv, acc[ct][2] * inv, acc[ct][3] * inv };
        v4f hi4 = { acc[ct][4] * inv, acc[ct][5] * inv, acc[ct][6] * inv, acc[ct][7] * inv };
        *(v4f*)(ob + ct * 16)     = lo;
        *(v4f*)(ob + ct * 16 + 4) = hi4;
    }
}

extern "C" void kernel_launch(void* const* d_in, const int* in_sizes, int n_in,
                              void* d_out, int out_size, void* d_ws, size_t ws_size,
                              hipStream_t stream) {
    (void)in_sizes; (void)n_in; (void)d_ws; (void)ws_size; (void)out_size;
    const float* x = (const float*)d_in[0];
    float* out = (float*)d_out;
    dim3 grid(NTOK / ROWS_PER_WG, NB);   // (64, 4) = 256 workgroups
    dim3 block(WAVES * 32);              // 128 threads = 4 wave32
    attn_f32_wmma_kernel<<<grid, block, 0, stream>>>(x, out);
}